// ButterflyLayer_82987358094151
// MI455X (gfx1250) — compile-verified
//
#include <hip/hip_runtime.h>

typedef float v2f __attribute__((ext_vector_type(2)));
typedef float v8f __attribute__((ext_vector_type(8)));

#define LROW     4096
#define NROWS    16
#define LDSPITCH 4100   // 4096 + 4: row stride = 16400B (16B aligned), ≡4 mod 64 banks

// -------------------------------------------------------------------------
// Kernel 1: compose each group of 4 butterfly stages into dense 16x16 mats.
// mats layout: [round][c][jin][jout]  (i.e. already transposed for WMMA B)
// Stage s pair (p, p+2^s) uses weights W[s][p][:] (top), W[s][p+2^s][:] (bot).
// -------------------------------------------------------------------------
__global__ void bfly_build_mats(const float* __restrict__ W,
                                float* __restrict__ mats)
{
    int idx   = blockIdx.x * blockDim.x + threadIdx.x;   // 0..12287
    int round = idx >> 12;
    int c     = (idx >> 4) & 255;
    int jin   = idx & 15;

    float v[16];
#pragma unroll
    for (int t = 0; t < 16; ++t) v[t] = (t == jin) ? 1.0f : 0.0f;

#pragma unroll
    for (int sp = 0; sp < 4; ++sp) {
        int s  = round * 4 + sp;
        int dd = 1 << sp;
        const float* Ws = W + (size_t)s * (LROW * 2);
#pragma unroll
        for (int t = 0; t < 16; ++t) {
            if (t & dd) continue;
            int tb = t + dd;
            int pt;
            if (round == 0)      pt = c * 16 + t;
            else if (round == 1) pt = (c >> 4) * 256 + t * 16 + (c & 15);
            else                 pt = t * 256 + (c >> 4) * 16 + (c & 15);
            int pb = pt + (1 << s);
            float top = v[t], bot = v[tb];
            v[t]  = Ws[2 * pt + 0] * top + Ws[2 * pt + 1] * bot;
            v[tb] = Ws[2 * pb + 0] * bot + Ws[2 * pb + 1] * top;
        }
    }

    float* dst = mats + (((size_t)round * 256 + c) * 16 + jin) * 16;
#pragma unroll
    for (int jout = 0; jout < 16; ++jout) dst[jout] = v[jout];
}

// -------------------------------------------------------------------------
// Kernel 2: fused 12-stage butterfly. 16 rows per block resident in LDS,
// 3 rounds of 256 dense 16x16 right-multiplies via V_WMMA_F32_16X16X4_F32.
// Bulk tile transfers use the CDNA5 async global<->LDS path (ASYNCcnt).
// -------------------------------------------------------------------------
__global__ __launch_bounds__(512) void bfly_main(const float* __restrict__ x,
                                                 const float* __restrict__ mats,
                                                 float* __restrict__ out)
{
    __shared__ float tile[NROWS * LDSPITCH];

    const int tid  = threadIdx.x;
    const int row0 = blockIdx.x * NROWS;

    // ---- async-load 16 rows (16384 x 16B by 512 lanes), no VGPR staging ----
#pragma unroll
    for (int i = 0; i < 32; ++i) {
        int lin = i * 512 + tid;
        int r   = lin >> 10;
        int c4  = lin & 1023;
        unsigned int  loff = (unsigned int)(uintptr_t)&tile[r * LDSPITCH + c4 * 4];
        const float*  g    = x + (size_t)(row0 + r) * LROW + c4 * 4;
        asm volatile("global_load_async_to_lds_b128 %0, %1, off"
                     :: "v"(loff), "v"(g) : "memory");
    }
    asm volatile("s_wait_asynccnt 0x0" ::: "memory");
    __syncthreads();

    const int lane = tid & 31;
    const int wave = tid >> 5;
    const int m    = lane & 15;          // M (row) for A; N (col) for B/D
    const int koff = (lane >> 4) << 1;   // upper half-lanes hold K+2 / M+8
    const int mhi  = (lane >> 4) * 8;

#pragma unroll
    for (int round = 0; round < 3; ++round) {
        const float* Mr = mats + (size_t)round * 256 * 256;

        for (int u = 0; u < 16; ++u) {
            int cb = wave * 16 + u;                 // this wave's 16x16x16 GEMM
            int base, stride;
            if (round == 0)      { base = cb * 16;                          stride = 1;   }
            else if (round == 1) { base = (cb >> 4) * 256 + (cb & 15);      stride = 16;  }
            else                 { base = (cb >> 4) * 16  + (cb & 15);      stride = 256; }

            const float* Bm   = Mr + (size_t)cb * 256;
            const float* arow = &tile[m * LDSPITCH + base];

            v8f acc = {0.f, 0.f, 0.f, 0.f, 0.f, 0.f, 0.f, 0.f};
#pragma unroll
            for (int kk = 0; kk < 4; ++kk) {
                int k0 = kk * 4 + koff;
                v2f a, b;
                a.x = arow[k0 * stride];
                a.y = arow[(k0 + 1) * stride];
                b.x = Bm[k0 * 16 + m];
                b.y = Bm[(k0 + 1) * 16 + m];
                acc = __builtin_amdgcn_wmma_f32_16x16x4_f32(
                        false, a, false, b, (short)0, acc, false, false);
            }

            // write D back (in place; this wave owns these 256 elements)
            float* drow = &tile[base + m * stride];
#pragma unroll
            for (int i = 0; i < 8; ++i)
                drow[(i + mhi) * LDSPITCH] = acc[i];
        }
        __syncthreads();
    }

    // ---- async-store 16 rows from LDS ----
#pragma unroll
    for (int i = 0; i < 32; ++i) {
        int lin = i * 512 + tid;
        int r   = lin >> 10;
        int c4  = lin & 1023;
        unsigned int loff = (unsigned int)(uintptr_t)&tile[r * LDSPITCH + c4 * 4];
        float*       g    = out + (size_t)(row0 + r) * LROW + c4 * 4;
        asm volatile("global_store_async_from_lds_b128 %0, %1, off"
                     :: "v"(g), "v"(loff) : "memory");
    }
    asm volatile("s_wait_asynccnt 0x0" ::: "memory");
}

extern "C" void kernel_launch(void* const* d_in, const int* in_sizes, int n_in,
                              void* d_out, int out_size, void* d_ws, size_t ws_size,
                              hipStream_t stream) {
    const float* x = (const float*)d_in[0];   // (8192, 4096) f32
    const float* W = (const float*)d_in[1];   // (12, 4096, 2) f32
    float* out  = (float*)d_out;              // (8192, 4096) f32
    float* mats = (float*)d_ws;               // needs 3*256*16*16*4 = 768 KB

    bfly_build_mats<<<48, 256, 0, stream>>>(W, mats);       // 12288 threads
    bfly_main<<<8192 / NROWS, 512, 0, stream>>>(x, mats, out);
}